// AlphaLayer_42691974922349
// MI455X (gfx1250) — compile-verified
//
#include <hip/hip_runtime.h>
#include <hip/hip_bf16.h>
#include <stdint.h>

// Problem constants (match reference: B=8192, N=4096)
#define BN   8192
#define NN   4096
#define MN   2048          // meeting point (N/2)
#define RPB  128           // rows per block (2 threads per row -> 256 thr)
#define BLK  256

typedef unsigned int u32x4 __attribute__((ext_vector_type(4)));
typedef int          i32x4 __attribute__((ext_vector_type(4)));
typedef int          i32x8 __attribute__((ext_vector_type(8)));
typedef float        f32x4 __attribute__((ext_vector_type(4)));

__device__ __forceinline__ float4 ld4(const float* p) {
    return *reinterpret_cast<const float4*>(p);
}
__device__ __forceinline__ void st4(float* p, float4 v) {
    *reinterpret_cast<float4*>(p) = v;
}
// Non-temporal 128-bit store (native ext-vector type: HIP float4 is rejected
// by __builtin_nontemporal_store).
__device__ __forceinline__ void st4_nt(float* p, float x, float y, float z, float w) {
    f32x4 v = { x, y, z, w };
    __builtin_nontemporal_store(v, reinterpret_cast<f32x4*>(p));
}

// v_rcp_f32 + one Newton-Raphson step (~1 ulp). Diagonal dominance of the
// system (b >= 5 > a+c) keeps denominators in [4.4, 6], so this is safe.
__device__ __forceinline__ float fast_rcp(float x) {
    float r = __builtin_amdgcn_rcpf(x);
    r = fmaf(r, fmaf(-x, r, 1.0f), r);
    return r;
}

// Downward elimination step (row i): carries cp, dp.
__device__ __forceinline__ void fstep(float xm, float x, float xp, float fi,
                                      float& cp, float& dp) {
    float a = xm * xm;
    float b = fmaxf(fmaf(x * x, x, 5.0f), 0.01f);
    float c = fmaf(xp, xp, xp + xp);
    float inv = fast_rcp(fmaf(-a, cp, b));
    cp = c * inv;
    dp = fmaf(-a, dp, fi) * inv;
}

// Upward elimination step (row j): carries e, g (u_j = g_j - e_j * u_{j-1}).
__device__ __forceinline__ void bstep(float xp, float x, float xm, float fi,
                                      float& ee, float& g) {
    float a = xm * xm;
    float b = fmaxf(fmaf(x * x, x, 5.0f), 0.01f);
    float c = fmaf(xp, xp, xp + xp);
    float inv = fast_rcp(fmaf(-c, ee, b));
    ee = a * inv;
    g = fmaf(-c, g, fi) * inv;
}

#if defined(__has_builtin)
#if __has_builtin(__builtin_amdgcn_tensor_load_to_lds) && __has_builtin(__builtin_amdgcn_s_wait_tensorcnt)
#define USE_TDM 1
#endif
#endif

__global__ __launch_bounds__(BLK)
void thomas_babe_kernel(const float* __restrict__ alpha,
                        const float* __restrict__ f,
                        float* __restrict__ out,     // dp/g then u
                        float* __restrict__ cpws) {  // cp/e scratch
    __shared__ __align__(16) float sf[NN];   // staged rhs vector (16 KB)
    __shared__ float sP[RPB];  // cp_{m-1} per row (from forward wave)
    __shared__ float sQ[RPB];  // dp_{m-1}
    __shared__ float sE[RPB];  // e_m      (from backward wave)
    __shared__ float sG[RPB];  // g_m

    const int tid  = threadIdx.x;
    const int wave = tid >> 5;
    const int lane = tid & 31;

#ifdef USE_TDM
    // ---- stage f into LDS via the Tensor Data Mover (one descriptor, wave 0) ----
    if (wave == 0) {
        const uint64_t ga  = (uint64_t)(uintptr_t)f;          // global byte addr (57-bit)
        const uint32_t lds = (uint32_t)(uintptr_t)&sf[0];     // LDS byte offset (addr[31:0])
        // D# group0: [1:0]=count=1 | [63:32]=lds_addr | [120:64]=global_addr | [127:126]=type=2
        u32x4 g0 = { 1u, lds, (uint32_t)ga, (uint32_t)(ga >> 32) | 0x80000000u };
        // D# group1: data_size=4B(code 2); tensor_dim0=4096; tensor_dim1=1;
        //            tile_dim0=4096; tile_dim1=1; tensor_dim0_stride=4096
        i32x8 g1 = { 0x00020000,              // wg_mask=0 | data_size=2<<16
                     0x10000000,              // tensor_dim0[15:0]=0x1000 << 16
                     0x00010000,              // tensor_dim0[31:16]=0 | tensor_dim1[15:0]=1 << 16
                     0x10000000,              // tensor_dim1[31:16]=0 | tile_dim0=0x1000 << 16
                     0x00000001,              // tile_dim1=1 | tile_dim2=0
                     0x00001000,              // tensor_dim0_stride[31:0]=4096
                     0, 0 };                  // stride hi / tensor_dim1_stride = 0
        i32x4 z4 = { 0, 0, 0, 0 };
#if __clang_major__ >= 23
        i32x8 z8 = { 0, 0, 0, 0, 0, 0, 0, 0 };
        __builtin_amdgcn_tensor_load_to_lds(g0, g1, z4, z4, z8, 0);
#else
        __builtin_amdgcn_tensor_load_to_lds(g0, g1, z4, z4, 0);
#endif
        __builtin_amdgcn_s_wait_tensorcnt(0);
    }
#else
    // ---- fallback: cooperative float4 staging ----
    for (int i = tid * 4; i < NN; i += BLK * 4) {
        st4(&sf[i], ld4(f + i));
    }
#endif
    __syncthreads();

    // Wave-uniform direction: even waves sweep down, odd waves sweep up.
    const int dir  = wave & 1;
    const int rloc = (wave >> 1) * 32 + lane;            // 0..127
    const long row = (long)blockIdx.x * RPB + rloc;

    const float* A  = alpha + row * (long)NN;
    float*       CP = cpws  + row * (long)NN;
    float*       DP = out   + row * (long)NN;

    // ================= Phase 1: twin eliminations =================
    if (dir == 0) {
        // rows i = 0 .. m-1
        float cp = 0.0f, dp = 0.0f, prev = 0.0f;  // alpha[-1] := 0 -> a_0 = 0
        float4 cur = ld4(A);
        for (int base = 0; base < MN; base += 4) {
            float4 nxt = ld4(A + base + 4);               // in-bounds up to A[MN+3]
            __builtin_prefetch(A + base + 68, 0, 0);      // global_prefetch_b8
            float4 cv, ov;
            fstep(prev,  cur.x, cur.y, sf[base + 0], cp, dp); cv.x = cp; ov.x = dp;
            fstep(cur.x, cur.y, cur.z, sf[base + 1], cp, dp); cv.y = cp; ov.y = dp;
            fstep(cur.y, cur.z, cur.w, sf[base + 2], cp, dp); cv.z = cp; ov.z = dp;
            fstep(cur.z, cur.w, nxt.x, sf[base + 3], cp, dp); cv.w = cp; ov.w = dp;
            st4(CP + base, cv);
            st4(DP + base, ov);
            prev = cur.w;
            cur  = nxt;
        }
        sP[rloc] = cp;   // cp_{m-1}
        sQ[rloc] = dp;   // dp_{m-1}
    } else {
        // rows j = N-1 .. m
        float ee = 0.0f, g = 0.0f, above = 0.0f;  // alpha[N] := 0 -> c_{N-1} = 0
        float4 cur = ld4(A + NN - 4);
        for (int base = NN - 4; base >= MN; base -= 4) {
            float4 nxt = ld4(A + base - 4);               // >= A[MN-4]
            __builtin_prefetch(A + base - 68, 0, 0);
            float4 ev, gv;
            bstep(above, cur.w, cur.z, sf[base + 3], ee, g); ev.w = ee; gv.w = g;
            bstep(cur.w, cur.z, cur.y, sf[base + 2], ee, g); ev.z = ee; gv.z = g;
            bstep(cur.z, cur.y, cur.x, sf[base + 1], ee, g); ev.y = ee; gv.y = g;
            bstep(cur.y, cur.x, nxt.w, sf[base + 0], ee, g); ev.x = ee; gv.x = g;
            st4(CP + base, ev);
            st4(DP + base, gv);
            above = cur.x;
            cur   = nxt;
        }
        sE[rloc] = ee;   // e_m
        sG[rloc] = g;    // g_m
    }
    __syncthreads();

    // ---- 2x2 middle coupling (each twin solves it redundantly) ----
    // u_m     = (g_m - e_m*dp_{m-1}) / (1 - e_m*cp_{m-1})
    // u_{m-1} = dp_{m-1} - cp_{m-1}*u_m
    const float cpm = sP[rloc], dpm = sQ[rloc];
    const float em  = sE[rloc], gm  = sG[rloc];
    const float um  = fmaf(-em, dpm, gm) * fast_rcp(fmaf(-em, cpm, 1.0f));

    // ================= Phase 2: twin back-substitutions =================
    // Final u is write-once/never-read: non-temporal stores keep the output
    // stream from evicting cp/dp lines (reversed-order reuse hits in L2).
    if (dir == 0) {
        float u = um;                                   // seed u_m
        for (int base = MN - 4; base >= 0; base -= 4) {
            float4 c4 = ld4(CP + base);
            float4 d4 = ld4(DP + base);
            float u3 = fmaf(-c4.w, u,  d4.w);
            float u2 = fmaf(-c4.z, u3, d4.z);
            float u1 = fmaf(-c4.y, u2, d4.y);
            float u0 = fmaf(-c4.x, u1, d4.x);
            st4_nt(DP + base, u0, u1, u2, u3);
            u = u0;
        }
    } else {
        float u = fmaf(-cpm, um, dpm);                  // seed u_{m-1}
        for (int base = MN; base < NN; base += 4) {
            float4 e4 = ld4(CP + base);
            float4 g4 = ld4(DP + base);
            float u0 = fmaf(-e4.x, u,  g4.x);
            float u1 = fmaf(-e4.y, u0, g4.y);
            float u2 = fmaf(-e4.z, u1, g4.z);
            float u3 = fmaf(-e4.w, u2, g4.w);
            st4_nt(DP + base, u0, u1, u2, u3);
            u = u3;
        }
    }
}

extern "C" void kernel_launch(void* const* d_in, const int* in_sizes, int n_in,
                              void* d_out, int out_size, void* d_ws, size_t ws_size,
                              hipStream_t stream) {
    (void)in_sizes; (void)n_in; (void)out_size; (void)ws_size;
    const float* alpha = (const float*)d_in[0];   // (B, N) fp32
    const float* f     = (const float*)d_in[1];   // (N,)   fp32
    float*       out   = (float*)d_out;           // (B, N) fp32
    float*       cpws  = (float*)d_ws;            // B*N fp32 scratch (128 MiB)

    dim3 grid(BN / RPB);   // 64 blocks
    dim3 block(BLK);       // 256 threads = 8 wave32s (4 fwd + 4 bwd)
    thomas_babe_kernel<<<grid, block, 0, stream>>>(alpha, f, out, cpws);
}